// HierarchicalEmergence_24515673325996
// MI455X (gfx1250) — compile-verified
//
#include <hip/hip_runtime.h>
#include <math.h>

typedef float v2f __attribute__((ext_vector_type(2)));
typedef float v8f __attribute__((ext_vector_type(8)));

#define N_CHILD   4096
#define N_PARENT  128
#define REG_EPS   1e-4f

// ---------------------------------------------------------------------------
// Kernel 1: Lambda[n] = inv(Sigma[n]) for 4096 SPD 16x16 matrices.
// One 256-thread block per matrix; Gauss-Jordan (SPD -> no pivoting needed).
// ---------------------------------------------------------------------------
__global__ void invert_sigma_kernel(const float* __restrict__ sigma,
                                    float* __restrict__ lambda) {
  __shared__ float A[16][17];
  __shared__ float B[16][17];
  const int n = blockIdx.x;
  const int t = threadIdx.x;
  const int r = t >> 4, c = t & 15;

  A[r][c] = sigma[(size_t)n * 256 + r * 16 + c];
  B[r][c] = (r == c) ? 1.f : 0.f;
  __syncthreads();

  for (int k = 0; k < 16; ++k) {
    float piv = A[k][k];
    __syncthreads();
    if (r == k) {
      float inv = 1.f / piv;
      A[k][c] *= inv;
      B[k][c] *= inv;
    }
    __syncthreads();
    float fac = A[r][k];
    float akc = A[k][c], bkc = B[k][c];
    __syncthreads();
    if (r != k) {
      A[r][c] -= fac * akc;
      B[r][c] -= fac * bkc;
    }
    __syncthreads();
  }
  lambda[(size_t)n * 256 + r * 16 + c] = B[r][c];
}

// ---------------------------------------------------------------------------
// Helpers for the rank-1 WMMA chain (vector kept in column 0 of a 16x16 tile)
// ---------------------------------------------------------------------------
// Extract column 0 of a C/D-layout tile into 16 wave-uniform scalars.
// C/D layout: lane 0 holds rows 0..7 (vgpr r), lane 16 holds rows 8..15.
__device__ __forceinline__ void extract_col0(float* vec, v8f C) {
#pragma unroll
  for (int j = 0; j < 8; ++j) {
    vec[j]     = __int_as_float(__builtin_amdgcn_readlane(__float_as_int(C[j]), 0));
    vec[j + 8] = __int_as_float(__builtin_amdgcn_readlane(__float_as_int(C[j]), 16));
  }
}

// Build B operand slice for rank-1 matrix vec*e0^T:
// B[kr][c] = vec[kr]*(c==0), kr = 4*kb + v + 2*hi.
__device__ __forceinline__ v2f masked_b(const float* vec, int kb, int c, int hi) {
  v2f b;
  b[0] = (c == 0) ? (hi ? vec[4 * kb + 2] : vec[4 * kb + 0]) : 0.f;
  b[1] = (c == 0) ? (hi ? vec[4 * kb + 3] : vec[4 * kb + 1]) : 0.f;
  return b;
}

// ---------------------------------------------------------------------------
// Kernel 2: one block (8 waves) per parent m.
//   accLp    = sum_n (w*T) * (Lambda * T^T)           (8 WMMA / child)
//   accInfoM = sum_n (w*T)*Lambda*T^T*T*(mu e0^T)     (16 WMMA / child, rank-1)
// then fused finalize: symmetrize, Jacobi eigh + clamp, Sigma = E D^-1 E^T,
// mu = Sigma * info.
// ---------------------------------------------------------------------------
__global__ void pool_kernel(const float* __restrict__ mu_children,  // (N,16)
                            const float* __restrict__ lambda,       // (N,16,16)
                            const float* __restrict__ weights,      // (N,M)
                            const float* __restrict__ transport,    // (M,N,16,16)
                            float* __restrict__ out) {              // mu(M,16) ++ sigma(M,16,16)
  const int m    = blockIdx.x;
  const int t    = threadIdx.x;
  const int lane = t & 31;
  const int wave = t >> 5;      // 8 waves of 32
  const int hi   = lane >> 4;   // half-wave id
  const int c    = lane & 15;   // column owned by this lane

  __shared__ float lpBuf[16][16];
  __shared__ float infoBuf[16];
  __shared__ float sA[16][17];
  __shared__ float sE[16][17];
  __shared__ float sS[16][17];

  lpBuf[t >> 4][t & 15] = 0.f;
  if (t < 16) infoBuf[t] = 0.f;
  __syncthreads();

  v8f accLp    = {0.f, 0.f, 0.f, 0.f, 0.f, 0.f, 0.f, 0.f};
  v8f accInfoM = {0.f, 0.f, 0.f, 0.f, 0.f, 0.f, 0.f, 0.f};

  for (int n = wave; n < N_CHILD; n += 8) {
    const float* T  = transport + ((size_t)m * N_CHILD + n) * 256;
    const float* L  = lambda + (size_t)n * 256;
    const float* MU = mu_children + (size_t)n * 16;
    const float  w  = weights[(size_t)n * N_PARENT + m];

    // Row fragments: lane holds row c, K = 4*kb + 2*hi + {0,1}  (float2 loads)
    // Column fragments tTc: lane holds T[k][c] (A operand of T^T), coalesced.
    v2f tT[4], tL[4], tMu[4], tTc[4];
#pragma unroll
    for (int kb = 0; kb < 4; ++kb) {
      const int k0 = 4 * kb + 2 * hi;
      tT[kb]     = *(const v2f*)(T + c * 16 + k0);
      tL[kb]     = *(const v2f*)(L + c * 16 + k0);
      tMu[kb]    = *(const v2f*)(MU + k0);
      tTc[kb][0] = T[(k0 + 0) * 16 + c];
      tTc[kb][1] = T[(k0 + 1) * 16 + c];
    }
    if (n + 8 < N_CHILD) __builtin_prefetch(T + 8 * 256, 0, 1);

    // w-scaled A fragment of T (shared by Lp path and info path)
    v2f aW[4];
#pragma unroll
    for (int kb = 0; kb < 4; ++kb) {
      aW[kb][0] = w * tT[kb][0];
      aW[kb][1] = w * tT[kb][1];
    }

    // ---- Lp path: V = Lambda * T^T, accLp += (w*T) * V ----
    v8f V = {0.f, 0.f, 0.f, 0.f, 0.f, 0.f, 0.f, 0.f};
#pragma unroll
    for (int kb = 0; kb < 4; ++kb) {
      V = __builtin_amdgcn_wmma_f32_16x16x4_f32(
          false, tL[kb], false, tT[kb], (short)0, V, false, false);
    }
#pragma unroll
    for (int kb = 0; kb < 4; ++kb) {
      // B operand from V's C/D layout: one half-wave swap per element
      v2f b;
#pragma unroll
      for (int v = 0; v < 2; ++v) {
        float bl, bh;
        if (kb < 2) {
          bl = V[4 * kb + v];
          bh = __shfl_xor(V[4 * kb + v + 2], 16, 32);
        } else {
          bl = __shfl_xor(V[4 * kb + v - 8], 16, 32);
          bh = V[4 * kb + v - 6];
        }
        b[v] = hi ? bh : bl;
      }
      accLp = __builtin_amdgcn_wmma_f32_16x16x4_f32(
          false, aW[kb], false, b, (short)0, accLp, false, false);
    }

    // ---- info path: rank-1 chain, vector lives in column 0 ----
    // C1 = T * (mu e0^T)  -> col0 = z = T*mu   (B frag is just masked tMu)
    v8f C1 = {0.f, 0.f, 0.f, 0.f, 0.f, 0.f, 0.f, 0.f};
#pragma unroll
    for (int kb = 0; kb < 4; ++kb) {
      v2f b;
      b[0] = (c == 0) ? tMu[kb][0] : 0.f;
      b[1] = (c == 0) ? tMu[kb][1] : 0.f;
      C1 = __builtin_amdgcn_wmma_f32_16x16x4_f32(
          false, tT[kb], false, b, (short)0, C1, false, false);
    }
    float zv[16];
    extract_col0(zv, C1);

    // C2 = T^T * (z e0^T) -> col0 = s = T^T*z
    v8f C2 = {0.f, 0.f, 0.f, 0.f, 0.f, 0.f, 0.f, 0.f};
#pragma unroll
    for (int kb = 0; kb < 4; ++kb) {
      C2 = __builtin_amdgcn_wmma_f32_16x16x4_f32(
          false, tTc[kb], false, masked_b(zv, kb, c, hi), (short)0, C2, false, false);
    }
    float sv[16];
    extract_col0(sv, C2);

    // C3 = Lambda * (s e0^T) -> col0 = q = Lambda*s
    v8f C3 = {0.f, 0.f, 0.f, 0.f, 0.f, 0.f, 0.f, 0.f};
#pragma unroll
    for (int kb = 0; kb < 4; ++kb) {
      C3 = __builtin_amdgcn_wmma_f32_16x16x4_f32(
          false, tL[kb], false, masked_b(sv, kb, c, hi), (short)0, C3, false, false);
    }
    float qv[16];
    extract_col0(qv, C3);

    // accInfoM += (w*T) * (q e0^T) ; col0 accumulates info contribution
#pragma unroll
    for (int kb = 0; kb < 4; ++kb) {
      accInfoM = __builtin_amdgcn_wmma_f32_16x16x4_f32(
          false, aW[kb], false, masked_b(qv, kb, c, hi), (short)0, accInfoM, false, false);
    }
  }

  // reduce the 8 waves through LDS float atomics (ds_add_f32)
#pragma unroll
  for (int r = 0; r < 8; ++r) atomicAdd(&lpBuf[r + 8 * hi][c], accLp[r]);
  if (c == 0) {
#pragma unroll
    for (int r = 0; r < 8; ++r) atomicAdd(&infoBuf[r + 8 * hi], accInfoM[r]);
  }
  __syncthreads();

  // ---- finalize: symmetrize, Jacobi eigh, clamp, Sigma = E D^-1 E^T ----
  {
    const int i = t >> 4, j = t & 15;
    sA[i][j] = 0.5f * (lpBuf[i][j] + lpBuf[j][i]);
    sE[i][j] = (i == j) ? 1.f : 0.f;
  }
  __syncthreads();

  for (int sweep = 0; sweep < 10; ++sweep) {
    for (int p = 0; p < 15; ++p) {
      for (int q = p + 1; q < 16; ++q) {
        const float app = sA[p][p], aqq = sA[q][q], apq = sA[p][q];
        float cr = 1.f, sr = 0.f;
        if (fabsf(apq) > 1e-12f) {
          const float tau = (aqq - app) / (2.f * apq);
          const float tt  = ((tau >= 0.f) ? 1.f : -1.f) /
                            (fabsf(tau) + sqrtf(1.f + tau * tau));
          cr = 1.f / sqrtf(1.f + tt * tt);
          sr = tt * cr;
        }
        float apj = 0.f, aqj = 0.f, ep = 0.f, eq = 0.f;
        if (t < 16)      { apj = sA[p][t]; aqj = sA[q][t]; }
        else if (t < 32) { ep = sE[t - 16][p]; eq = sE[t - 16][q]; }
        __syncthreads();
        if (t < 16) {
          if (t == p) {
            sA[p][p] = cr * cr * app - 2.f * cr * sr * apq + sr * sr * aqq;
          } else if (t == q) {
            sA[q][q] = sr * sr * app + 2.f * cr * sr * apq + cr * cr * aqq;
            sA[p][q] = 0.f;
            sA[q][p] = 0.f;
          } else {
            const float np = cr * apj - sr * aqj;
            const float nq = sr * apj + cr * aqj;
            sA[p][t] = np; sA[t][p] = np;
            sA[q][t] = nq; sA[t][q] = nq;
          }
        } else if (t < 32) {
          const int i = t - 16;
          sE[i][p] = cr * ep - sr * eq;
          sE[i][q] = sr * ep + cr * eq;
        }
        __syncthreads();
      }
    }
  }

  // Sigma = E * diag(1/max(lambda, eps)) * E^T   (clamp == reference eigh path)
  {
    const int i = t >> 4, j = t & 15;
    float s = 0.f;
#pragma unroll
    for (int k = 0; k < 16; ++k) {
      float lam = sA[k][k];
      lam = (lam > REG_EPS) ? lam : REG_EPS;
      s += sE[i][k] * sE[j][k] / lam;
    }
    sS[i][j] = s;
    out[2048 + (size_t)m * 256 + i * 16 + j] = s;   // sigma_pooled
  }
  __syncthreads();
  if (t < 16) {
    float s = 0.f;
#pragma unroll
    for (int j = 0; j < 16; ++j) s += sS[t][j] * infoBuf[j];
    out[(size_t)m * 16 + t] = s;                    // mu_pooled
  }
}

// ---------------------------------------------------------------------------
extern "C" void kernel_launch(void* const* d_in, const int* in_sizes, int n_in,
                              void* d_out, int out_size, void* d_ws, size_t ws_size,
                              hipStream_t stream) {
  const float* mu        = (const float*)d_in[0];  // (4096,16)
  const float* sigma     = (const float*)d_in[1];  // (4096,16,16)
  const float* weights   = (const float*)d_in[2];  // (4096,128)
  const float* transport = (const float*)d_in[3];  // (128,4096,16,16)
  float*       out       = (float*)d_out;          // 2048 + 32768 floats
  float*       lambda    = (float*)d_ws;           // 4096*256 floats = 4 MB

  invert_sigma_kernel<<<N_CHILD, 256, 0, stream>>>(sigma, lambda);
  pool_kernel<<<N_PARENT, 256, 0, stream>>>(mu, lambda, weights, transport, out);
}